// DetectUDPModel_12876311953925
// MI455X (gfx1250) — compile-verified
//
#include <hip/hip_runtime.h>
#include <hip/hip_bf16.h>
#include <math.h>

typedef float v2f __attribute__((ext_vector_type(2)));
typedef float v8f __attribute__((ext_vector_type(8)));

#define NDIM 128
#define MROWS 2048
#define KTOT (NDIM * NDIM)        // 16384
#define KSPLITS 16
#define WAVES_PER_BLOCK 8
#define MTILES (MROWS / 16)       // 128
// per-wave K range: KTOT/KSPLITS/WAVES_PER_BLOCK = 128 floats -> 32 chunks of 4
#define ITERS (KTOT / KSPLITS / WAVES_PER_BLOCK / 4)   // 32

// ---------------------------------------------------------------------------
// Kernel 1: build xr/xi = Re/Im(matH.flatten()) from theta (4,128) and evl (2)
// ---------------------------------------------------------------------------
__global__ __launch_bounds__(128) void build_x_kernel(
    const float* __restrict__ theta, const float* __restrict__ evl,
    float* __restrict__ xr, float* __restrict__ xi) {
  __shared__ float s0r[NDIM], s0i[NDIM], s1r[NDIM], s1i[NDIM];
  const int t = threadIdx.x;
  s0r[t] = theta[0 * NDIM + t];
  s0i[t] = theta[1 * NDIM + t];
  s1r[t] = theta[2 * NDIM + t];
  s1i[t] = theta[3 * NDIM + t];
  __syncthreads();

  float n0 = 0.f, dr = 0.f, di = 0.f;
  for (int u = 0; u < NDIM; ++u) {
    n0 += s0r[u] * s0r[u] + s0i[u] * s0i[u];
    dr += s0r[u] * s1r[u] + s0i[u] * s1i[u];   // Re(conj(c0)*c1)
    di += s0r[u] * s1i[u] - s0i[u] * s1r[u];   // Im(conj(c0)*c1)
  }
  const float inv0 = 1.0f / sqrtf(n0);
  dr *= inv0;
  di *= inv0;

  const float c1pr = s1r[t] - inv0 * (dr * s0r[t] - di * s0i[t]);
  const float c1pi = s1i[t] - inv0 * (dr * s0i[t] + di * s0r[t]);
  __syncthreads();
  s1r[t] = c1pr;
  s1i[t] = c1pi;
  __syncthreads();

  float n1 = 0.f;
  for (int u = 0; u < NDIM; ++u) n1 += s1r[u] * s1r[u] + s1i[u] * s1i[u];
  const float inv1 = 1.0f / sqrtf(n1);

  float l0 = log1pf(expf(evl[0]));
  float l1 = log1pf(expf(evl[1]));
  const float ln = 1.0f / sqrtf(l0 * l0 + l1 * l1);
  l0 *= ln;
  l1 *= ln;

  const float f0 = l0 * inv0 * inv0;
  const float f1 = l1 * inv1 * inv1;
  const float ar = s0r[t], ai = s0i[t];
  const float cr = s1r[t], ci = s1i[t];
  for (int b = 0; b < NDIM; ++b) {
    const float br = s0r[b], bi = s0i[b];
    const float er = s1r[b], ei = s1i[b];
    xr[t * NDIM + b] = f0 * (ar * br + ai * bi) - f1 * (cr * er + ci * ei);
    xi[t * NDIM + b] = f0 * (ai * br - ar * bi) - f1 * (ci * er - cr * ei);
  }
}

// ---------------------------------------------------------------------------
// Kernel 2: bandwidth-bound complex matvec via V_WMMA_F32_16X16X4_F32.
// SGPR-base + 32-bit VGPR byte-offset addressing; explicit register
// double-buffering so each chunk's loads are issued one iteration ahead.
// ---------------------------------------------------------------------------
__global__ __launch_bounds__(256) void matvec_wmma_kernel(
    const float* __restrict__ re, const float* __restrict__ im,
    const float* __restrict__ xr, const float* __restrict__ xi,
    float* __restrict__ partial) {
  __shared__ float lre[WAVES_PER_BLOCK][16];
  __shared__ float lim[WAVES_PER_BLOCK][16];

  const int lane = threadIdx.x & 31;
  const int wave = threadIdx.x >> 5;
  const int half = lane >> 4;    // K-pair select: 0 -> K0,K1 ; 1 -> K2,K3
  const int mrow = lane & 15;    // row within the 16-row tile
  const int mtile = blockIdx.x;
  const int ksplit = blockIdx.y;

  // contiguous per-wave K range (16 B per iteration per lane)
  const unsigned kstart =
      (unsigned)(ksplit * (KTOT / KSPLITS) + wave * (KTOT / KSPLITS / WAVES_PER_BLOCK));
  // 32-bit BYTE offsets (max < 2^31) against uniform SGPR bases
  unsigned offA = ((unsigned)(mtile * 16 + mrow) * KTOT + kstart + 2u * half) * 4u;
  unsigned offX = (kstart + 2u * half) * 4u;

  const char* reB = (const char*)re;
  const char* imB = (const char*)im;
  const char* xrB = (const char*)xr;
  const char* xiB = (const char*)xi;

  v8f acc0 = {};  // Re*xr + Im*xi  -> v_re
  v8f acc1 = {};  // Re*xi
  v8f acc2 = {};  // Im*xr          -> v_im = acc1 - acc2

  v2f are = *(const v2f*)(reB + offA);
  v2f aim = *(const v2f*)(imB + offA);
  v2f bxr = *(const v2f*)(xrB + offX);
  v2f bxi = *(const v2f*)(xiB + offX);

#pragma unroll 4
  for (int i = 0; i < ITERS - 1; ++i) {
    offA += 16u;  // 4 floats per chunk
    offX += 16u;
    const v2f nare = *(const v2f*)(reB + offA);
    const v2f naim = *(const v2f*)(imB + offA);
    const v2f nbxr = *(const v2f*)(xrB + offX);
    const v2f nbxi = *(const v2f*)(xiB + offX);
    acc0 = __builtin_amdgcn_wmma_f32_16x16x4_f32(false, are, false, bxr,
                                                 (short)0, acc0, false, false);
    acc0 = __builtin_amdgcn_wmma_f32_16x16x4_f32(false, aim, false, bxi,
                                                 (short)0, acc0, false, false);
    acc1 = __builtin_amdgcn_wmma_f32_16x16x4_f32(false, are, false, bxi,
                                                 (short)0, acc1, false, false);
    acc2 = __builtin_amdgcn_wmma_f32_16x16x4_f32(false, aim, false, bxr,
                                                 (short)0, acc2, false, false);
    are = nare;
    aim = naim;
    bxr = nbxr;
    bxi = nbxi;
  }
  // peeled last chunk (no out-of-bounds prefetch)
  acc0 = __builtin_amdgcn_wmma_f32_16x16x4_f32(false, are, false, bxr,
                                               (short)0, acc0, false, false);
  acc0 = __builtin_amdgcn_wmma_f32_16x16x4_f32(false, aim, false, bxi,
                                               (short)0, acc0, false, false);
  acc1 = __builtin_amdgcn_wmma_f32_16x16x4_f32(false, are, false, bxi,
                                               (short)0, acc1, false, false);
  acc2 = __builtin_amdgcn_wmma_f32_16x16x4_f32(false, aim, false, bxr,
                                               (short)0, acc2, false, false);

  // C/D layout: VGPR r of lane l holds row (r + 8*(l>=16)), col l%16; all
  // columns are identical, so lanes 0 and 16 carry the 16 row sums.
  if (mrow == 0) {
#pragma unroll
    for (int r = 0; r < 8; ++r) {
      lre[wave][8 * half + r] = acc0[r];
      lim[wave][8 * half + r] = acc1[r] - acc2[r];
    }
  }
  __syncthreads();

  if (threadIdx.x < 32) {
    const int t = threadIdx.x;
    float s = 0.f;
    if (t < 16) {
      for (int w = 0; w < WAVES_PER_BLOCK; ++w) s += lre[w][t];
    } else {
      for (int w = 0; w < WAVES_PER_BLOCK; ++w) s += lim[w][t - 16];
    }
    partial[(mtile * KSPLITS + ksplit) * 32 + t] = s;
  }
}

// ---------------------------------------------------------------------------
// Kernel 3: deterministic finalize — sum K-split partials per row, then
// sum |v|^2 over all 2048 rows with a single-block tree reduction.
// ---------------------------------------------------------------------------
__global__ __launch_bounds__(256) void finalize_kernel(
    const float* __restrict__ partial, float* __restrict__ out) {
  __shared__ float red[256];
  const int t = threadIdx.x;
  float acc = 0.f;
  for (int rr = 0; rr < MROWS / 256; ++rr) {
    const int row = t * (MROWS / 256) + rr;
    const int mtile = row >> 4;
    const int rloc = row & 15;
    const float* p = partial + mtile * KSPLITS * 32;
    float vre = 0.f, vim = 0.f;
    for (int ks = 0; ks < KSPLITS; ++ks) {
      vre += p[ks * 32 + rloc];
      vim += p[ks * 32 + 16 + rloc];
    }
    acc += vre * vre + vim * vim;
  }
  red[t] = acc;
  __syncthreads();
  for (int s = 128; s > 0; s >>= 1) {
    if (t < s) red[t] += red[t + s];
    __syncthreads();
  }
  if (t == 0) out[0] = red[0];
}

// ---------------------------------------------------------------------------
extern "C" void kernel_launch(void* const* d_in, const int* in_sizes, int n_in,
                              void* d_out, int out_size, void* d_ws,
                              size_t ws_size, hipStream_t stream) {
  const float* basis_re = (const float*)d_in[0];
  const float* basis_im = (const float*)d_in[1];
  const float* theta = (const float*)d_in[2];
  const float* evl = (const float*)d_in[3];

  float* xr = (float*)d_ws;                  // 16384 floats
  float* xi = xr + KTOT;                     // 16384 floats
  float* partial = xi + KTOT;                // 128*16*32 = 65536 floats

  build_x_kernel<<<1, 128, 0, stream>>>(theta, evl, xr, xi);

  dim3 grid(MTILES, KSPLITS);
  matvec_wmma_kernel<<<grid, 32 * WAVES_PER_BLOCK, 0, stream>>>(
      basis_re, basis_im, xr, xi, partial);

  finalize_kernel<<<1, 256, 0, stream>>>(partial, (float*)d_out);
}